// IntVOS_27015344292335
// MI455X (gfx1250) — compile-verified
//
#include <hip/hip_runtime.h>

#define NPTS   9216
#define DIM    128
#define NTILES 576   // 9216 / 16

typedef __attribute__((ext_vector_type(2))) float v2f;
typedef __attribute__((ext_vector_type(8))) float v8f;

// Branchless insert of d into ascending sorted triple (a0 <= a1 <= a2),
// keeping the 3 smallest values seen so far.
__device__ __forceinline__ void ins3(float& a0, float& a1, float& a2, float d) {
  a2 = fminf(a2, fmaxf(a1, d));   // uses old a1
  a1 = fminf(a1, fmaxf(a0, d));   // uses old a0
  a0 = fminf(a0, d);
}

// One wave per row: squared L2 norms of reference (rows 0..NPTS-1) and query
// (rows NPTS..2*NPTS-1) embeddings.
__global__ __launch_bounds__(256) void norms_kernel(const float* __restrict__ r,
                                                    const float* __restrict__ q,
                                                    float* __restrict__ rs,
                                                    float* __restrict__ qs) {
  const int row  = blockIdx.x * 8 + (threadIdx.x >> 5);
  const int lane = threadIdx.x & 31;
  const float* src = (row < NPTS) ? (r + (size_t)row * DIM)
                                  : (q + (size_t)(row - NPTS) * DIM);
  const float4 v = *(const float4*)(src + lane * 4);
  float s = v.x * v.x + v.y * v.y + v.z * v.z + v.w * v.w;
  #pragma unroll
  for (int off = 16; off > 0; off >>= 1) s += __shfl_xor(s, off, 32);
  if (lane == 0) {
    if (row < NPTS) rs[row] = s;
    else            qs[row - NPTS] = s;
  }
}

// Fused: D = rs - 2*(R @ Q^T) via f32 WMMA, running top-3 per (query, object),
// masked-mean feature like the reference.
__global__ __launch_bounds__(256) void knn_wmma_kernel(
    const float* __restrict__ r, const float* __restrict__ q,
    const int* __restrict__ labels, const float* __restrict__ rs,
    const float* __restrict__ qs, const int* __restrict__ kptr,
    float* __restrict__ out) {
  const int wave  = threadIdx.x >> 5;
  const int lane  = threadIdx.x & 31;
  const int lq    = lane & 15;   // matrix row/col within half-wave
  const int kh    = lane >> 4;   // which K/M half this lane covers
  const int qbase = blockIdx.x * 16;
  const float INFV = 1e30f;

  // Preload query (B-matrix) fragments for all 32 K-steps: per lane, query row
  // (qbase+lq), columns {4*kk + 2*kh, +1}.  64 VGPRs, reused for every tile.
  v2f qv[32];
  {
    const float* qp = q + (size_t)(qbase + lq) * DIM + 2 * kh;
    #pragma unroll
    for (int kk = 0; kk < 32; ++kk) qv[kk] = *(const v2f*)(qp + 4 * kk);
  }

  // top-3 (ascending) per object; INF sentinel == "invalid / not found"
  float t00=INFV,t01=INFV,t02=INFV;
  float t10=INFV,t11=INFV,t12=INFV;
  float t20=INFV,t21=INFV,t22=INFV;

  for (int ct = wave; ct < NTILES; ct += 8) {
    const int nbase = ct << 4;
    const float* rp = r + (size_t)(nbase + lq) * DIM + 2 * kh;

    // ---- Phase 1: issue ALL tile loads up front (A fragments + labels + ||r||^2)
    // Loads return in order, so the kk-th WMMA below only needs the kk-th load;
    // the full load latency overlaps the WMMA stream.
    v2f av[32];
    #pragma unroll
    for (int kk = 0; kk < 32; ++kk) av[kk] = *(const v2f*)(rp + 4 * kk);

    const int rrow = nbase + 8 * kh;
    int lab[8]; float rn[8];
    #pragma unroll
    for (int j = 0; j < 8; ++j) { lab[j] = labels[rrow + j]; rn[j] = rs[rrow + j]; }

    // Prefetch next tile this wave will touch (global_prefetch_b8).
    if (ct + 8 < NTILES) {
      const float* rpn = r + (size_t)(((ct + 8) << 4) + lq) * DIM;
      __builtin_prefetch(rpn, 0, 3);
    }

    // ---- Phase 2: 32 back-to-back f32 WMMAs (K = 128 in steps of 4)
    v8f c = {};
    #pragma unroll
    for (int kk = 0; kk < 32; ++kk) {
      c = __builtin_amdgcn_wmma_f32_16x16x4_f32(
              false, av[kk], false, qv[kk], (short)0, c, false, false);
    }

    // ---- Phase 3: top-3 maintenance.
    // C layout: c[j] = dot(r[nbase + j + 8*kh], q[qbase + lq])
    #pragma unroll
    for (int j = 0; j < 8; ++j) {
      const float d  = fmaf(-2.0f, c[j], rn[j]);   // rs - 2*q.r  (qs added later)
      const float d0 = (lab[j] == 0) ? d : INFV;
      const float d1 = (lab[j] == 1) ? d : INFV;
      const float d2 = (lab[j] == 2) ? d : INFV;
      ins3(t00,t01,t02,d0);
      ins3(t10,t11,t12,d1);
      ins3(t20,t21,t22,d2);
    }
  }

  // Merge lane L with lane L^16 (same query, other reference half).
  {
    float p00=__shfl_xor(t00,16,32), p01=__shfl_xor(t01,16,32), p02=__shfl_xor(t02,16,32);
    float p10=__shfl_xor(t10,16,32), p11=__shfl_xor(t11,16,32), p12=__shfl_xor(t12,16,32);
    float p20=__shfl_xor(t20,16,32), p21=__shfl_xor(t21,16,32), p22=__shfl_xor(t22,16,32);
    ins3(t00,t01,t02,p00); ins3(t00,t01,t02,p01); ins3(t00,t01,t02,p02);
    ins3(t10,t11,t12,p10); ins3(t10,t11,t12,p11); ins3(t10,t11,t12,p12);
    ins3(t20,t21,t22,p20); ins3(t20,t21,t22,p21); ins3(t20,t21,t22,p22);
  }

  // Cross-wave merge through LDS.
  __shared__ float sm[16][8][9];
  if (kh == 0) {
    float* s = &sm[lq][wave][0];
    s[0]=t00; s[1]=t01; s[2]=t02;
    s[3]=t10; s[4]=t11; s[5]=t12;
    s[6]=t20; s[7]=t21; s[8]=t22;
  }
  __syncthreads();

  if (threadIdx.x < 16) {
    const int qi = threadIdx.x;
    float f[9];
    #pragma unroll
    for (int j = 0; j < 9; ++j) f[j] = sm[qi][0][j];
    for (int w = 1; w < 8; ++w) {
      #pragma unroll
      for (int tr = 0; tr < 3; ++tr) {
        #pragma unroll
        for (int i = 0; i < 3; ++i) {
          ins3(f[tr*3+0], f[tr*3+1], f[tr*3+2], sm[qi][w][tr*3+i]);
        }
      }
    }
    const float qn = qs[qbase + qi];
    const int   k  = *kptr;
    const float VLIM = 1e29f;
    #pragma unroll
    for (int obj = 0; obj < 3; ++obj) {
      float a = f[obj*3], b = f[obj*3+1], cc = f[obj*3+2];
      bool v0 = a < VLIM, v1 = b < VLIM, v2 = cc < VLIM;
      float res;
      if (!v0) {
        res = (k == 1) ? 1.0e20f : 0.0f;        // no valid neighbor at all
      } else if (k == 1) {
        res = a + qn;
      } else {
        // invalid entries replaced by max valid entry, then mean (ref semantics)
        float mx = v2 ? cc : (v1 ? b : a);
        float bb = v1 ? b : mx;
        float c2 = v2 ? cc : mx;
        res = (a + bb + c2) * (1.0f / 3.0f) + qn;
      }
      out[(size_t)(qbase + qi) * 3 + obj] = res;
    }
  }
}

extern "C" void kernel_launch(void* const* d_in, const int* in_sizes, int n_in,
                              void* d_out, int out_size, void* d_ws, size_t ws_size,
                              hipStream_t stream) {
  (void)in_sizes; (void)n_in; (void)out_size; (void)ws_size;
  const float* r      = (const float*)d_in[0];   // reference_embeddings [9216,128]
  const float* q      = (const float*)d_in[1];   // query_embeddings     [9216,128]
  const int*   labels = (const int*)d_in[2];     // reference_labels     [9216]
  const int*   kptr   = (const int*)d_in[3];     // k_nearest_neighbors  scalar
  float*       out    = (float*)d_out;           // [9216*3]
  float*       rs     = (float*)d_ws;            // ||r||^2  [9216]
  float*       qs     = rs + NPTS;               // ||q||^2  [9216]

  norms_kernel<<<(2 * NPTS) / 8, 256, 0, stream>>>(r, q, rs, qs);
  knn_wmma_kernel<<<NTILES, 256, 0, stream>>>(r, q, labels, rs, qs, kptr, out);
}